// HebbianLayer_755914244754
// MI455X (gfx1250) — compile-verified
//
#include <hip/hip_runtime.h>

#define IN_DIM  8192
#define OUT_DIM 4096
#define ALPHA   0.001f
#define GAMMA   0.99f

typedef float v2f __attribute__((ext_vector_type(2)));
typedef float v8f __attribute__((ext_vector_type(8)));

// split-K for the WMMA GEMV
#define KSLICES 8
#define KS_LEN  (IN_DIM / KSLICES)      // 1024

// row-block scan decomposition
#define RB       32
#define RB_LEN   (OUT_DIM / RB)         // 128 rows per block
#define CB_THREADS 256
#define CBLOCKS  (IN_DIM / CB_THREADS)  // 32 column blocks

// ---------------------------------------------------------------------------
// Kernel 1: pre-partials via V_WMMA_F32_16X16X4_F32.
// One wave handles 16 rows of W over a 1024-wide K slice.
// A (16x4 f32): lanes 0-15 -> M=0..15, VGPR0=K0,VGPR1=K1; lanes 16-31 -> K2,K3.
// B (4x16 f32): broadcast x[k..k+3] to all 16 columns (same per-lane K layout).
// Every column of D equals the row dot; extract N=0 from lane 0 (M=0..7) and
// lane 16 (M=8..15).
// ---------------------------------------------------------------------------
__global__ void __launch_bounds__(32)
k_gemv_wmma(const float* __restrict__ W, const float* __restrict__ x,
            float* __restrict__ pre_part) {
  const int lane = threadIdx.x;
  const int hf   = lane >> 4;      // which K pair this lane supplies
  const int m    = lane & 15;      // row within the 16-row tile
  const int m0   = blockIdx.x * 16;
  const int ks   = blockIdx.y;
  const float* wrow = W + (size_t)(m0 + m) * IN_DIM;

  v8f c = {};
  const int k0 = ks * KS_LEN;
  for (int k = k0; k < k0 + KS_LEN; k += 4) {
    const int kk = k + 2 * hf;
    v2f a; a.x = wrow[kk]; a.y = wrow[kk + 1];
    v2f b; b.x = x[kk];    b.y = x[kk + 1];
    c = __builtin_amdgcn_wmma_f32_16x16x4_f32(
        /*neg_a=*/false, a, /*neg_b=*/false, b,
        /*c_mod=*/(short)0, c, /*reuse_a=*/false, /*reuse_b=*/false);
  }

  float* dst = pre_part + (size_t)ks * OUT_DIM + m0;
  if (lane == 0) {
#pragma unroll
    for (int r = 0; r < 8; ++r) dst[r] = c[r];
  } else if (lane == 16) {
#pragma unroll
    for (int r = 0; r < 8; ++r) dst[8 + r] = c[r];
  }
}

// ---------------------------------------------------------------------------
// Kernel 2: reduce split-K partials, relu, global max, y = r^2 / max^2,
// EMA of activities. Single 1024-thread block (4 rows/thread).
// ---------------------------------------------------------------------------
__global__ void __launch_bounds__(1024)
k_activate(const float* __restrict__ pre_part, const float* __restrict__ exp_avg,
           float* __restrict__ y_ws, float* __restrict__ out_y,
           float* __restrict__ out_ea) {
  __shared__ float red[1024];
  const int t = threadIdx.x;
  float r[OUT_DIM / 1024];
  float lmax = 0.0f;
#pragma unroll
  for (int i = 0; i < OUT_DIM / 1024; ++i) {
    const int row = t + i * 1024;
    float s = 0.0f;
#pragma unroll
    for (int p = 0; p < KSLICES; ++p) s += pre_part[p * OUT_DIM + row];
    const float rv = s > 0.0f ? s : 0.0f;   // relu
    r[i] = rv;
    lmax = fmaxf(lmax, rv);
  }
  red[t] = lmax;
  __syncthreads();
  for (int off = 512; off > 0; off >>= 1) {
    if (t < off) red[t] = fmaxf(red[t], red[t + off]);
    __syncthreads();
  }
  const float mx  = red[0];
  const float inv = 1.0f / (mx * mx);       // LAMB == 2
#pragma unroll
  for (int i = 0; i < OUT_DIM / 1024; ++i) {
    const int row = t + i * 1024;
    const float yv = r[i] * r[i] * inv;
    y_ws[row]  = yv;
    out_y[row] = yv;
    out_ea[row] = GAMMA * exp_avg[row] + (1.0f - GAMMA) * yv;
  }
}

// ---------------------------------------------------------------------------
// Kernel 3 (scan pass A): per-rowblock partial sums S[rb][j] = sum y[k]*W[k,j]
// over a 128-row block. Columns are thread-parallel (fully coalesced); the
// y tile is staged in LDS. W should hit L2 (loaded by kernel 1).
// ---------------------------------------------------------------------------
__global__ void __launch_bounds__(CB_THREADS)
k_partial(const float* __restrict__ W, const float* __restrict__ y,
          float* __restrict__ S) {
  __shared__ float ys[RB_LEN];
  const int t  = threadIdx.x;
  const int j  = blockIdx.x * CB_THREADS + t;
  const int rb = blockIdx.y;
  const int k0 = rb * RB_LEN;
  if (t < RB_LEN) ys[t] = y[k0 + t];
  __syncthreads();

  float s = 0.0f;
  const float* Wp = W + (size_t)k0 * IN_DIM + j;
#pragma unroll 4
  for (int i = 0; i < RB_LEN; ++i) s += ys[i] * Wp[(size_t)i * IN_DIM];
  S[(size_t)rb * IN_DIM + j] = s;
}

// ---------------------------------------------------------------------------
// Kernel 4 (scan pass B + apply): fold the exclusive block offset, then walk
// the 128-row block maintaining the inclusive running sum and emit
//   new_W[k,j] = W[k,j] + ALPHA*(y[k]*x[j] - y[k]*cumsum)
// W read is last-use (non-temporal load); new_W stream is non-temporal store
// so it does not evict W from L2.
// ---------------------------------------------------------------------------
__global__ void __launch_bounds__(CB_THREADS)
k_scan_update(const float* __restrict__ W, const float* __restrict__ x,
              const float* __restrict__ y, const float* __restrict__ S,
              float* __restrict__ Wout) {
  __shared__ float ys[RB_LEN];
  const int t  = threadIdx.x;
  const int j  = blockIdx.x * CB_THREADS + t;
  const int rb = blockIdx.y;
  const int k0 = rb * RB_LEN;
  if (t < RB_LEN) ys[t] = y[k0 + t];
  __syncthreads();

  float s = 0.0f;                       // exclusive prefix over row blocks
  for (int b = 0; b < rb; ++b) s += S[(size_t)b * IN_DIM + j];

  const float xj = x[j];
  const float* Wp = W    + (size_t)k0 * IN_DIM + j;
  float*       Op = Wout + (size_t)k0 * IN_DIM + j;
#pragma unroll 4
  for (int i = 0; i < RB_LEN; ++i) {
    const float w  = __builtin_nontemporal_load(&Wp[(size_t)i * IN_DIM]);
    const float yk = ys[i];
    s += yk * w;                        // inclusive cumsum of y*W down rows
    const float nw = fmaf(ALPHA, fmaf(yk, xj, -(yk * s)), w);
    __builtin_nontemporal_store(nw, &Op[(size_t)i * IN_DIM]);
  }
}

// ---------------------------------------------------------------------------
extern "C" void kernel_launch(void* const* d_in, const int* in_sizes, int n_in,
                              void* d_out, int out_size, void* d_ws,
                              size_t ws_size, hipStream_t stream) {
  const float* x  = (const float*)d_in[0];
  const float* W  = (const float*)d_in[1];
  const float* ea = (const float*)d_in[2];

  float* out    = (float*)d_out;
  float* out_y  = out;                                      // [4096]
  float* out_W  = out + OUT_DIM;                            // [4096*8192]
  float* out_ea = out + OUT_DIM + (size_t)OUT_DIM * IN_DIM; // [4096]

  float* ws       = (float*)d_ws;
  float* pre_part = ws;                            // [KSLICES][OUT_DIM]
  float* y_ws     = pre_part + KSLICES * OUT_DIM;  // [OUT_DIM]
  float* S        = y_ws + OUT_DIM;                // [RB][IN_DIM]

  k_gemv_wmma<<<dim3(OUT_DIM / 16, KSLICES), 32, 0, stream>>>(W, x, pre_part);
  k_activate<<<1, 1024, 0, stream>>>(pre_part, ea, y_ws, out_y, out_ea);
  k_partial<<<dim3(CBLOCKS, RB), CB_THREADS, 0, stream>>>(W, y_ws, S);
  k_scan_update<<<dim3(CBLOCKS, RB), CB_THREADS, 0, stream>>>(W, x, y_ws, S, out_W);
}